// BayesianLayer_64742337020054
// MI455X (gfx1250) — compile-verified
//
#include <hip/hip_runtime.h>

// Problem dims (fixed by reference)
#define NB   128
#define NIN  1024
#define NOUT 1024

typedef __attribute__((ext_vector_type(2))) float v2f;
typedef __attribute__((ext_vector_type(4))) float v4f;
typedef __attribute__((ext_vector_type(8))) float v8f;

__device__ __forceinline__ float softplus1(float v) {
    // numerically stable: max(v,0) + log1p(exp(-|v|))
    return fmaxf(v, 0.0f) + log1pf(expf(-fabsf(v)));
}

__device__ __forceinline__ v4f softplus4(v4f r) {
    v4f s;
    s.x = softplus1(r.x);
    s.y = softplus1(r.y);
    s.z = softplus1(r.z);
    s.w = softplus1(r.w);
    return s;
}

// ---------------------------------------------------------------------------
// Kernel 1: sigma_ws[i,o] = softplus(ro[i,o])   (1M elems, 4 per thread)
// ---------------------------------------------------------------------------
__global__ void sigma_kernel(const float* __restrict__ ro, float* __restrict__ sigma) {
    size_t idx = ((size_t)blockIdx.x * blockDim.x + threadIdx.x) * 4;
    v4f r = *(const v4f*)(ro + idx);
    *(v4f*)(sigma + idx) = softplus4(r);
}

// ---------------------------------------------------------------------------
// Kernel 2: out = x @ mu via V_WMMA_F32_16X16X4_F32.
// One wave per 16x16 tile of out. M=128, N=1024, K=1024.
// Two independent accumulators pipeline past the WMMA->WMMA RAW hazard.
//
// VGPR layouts per CDNA5 ISA 7.12.2:
//   A 16x4:  lane<16 holds M=lane, {K=k+0,k+1}; lane>=16 holds M=lane-16, {K=k+2,k+3}
//   B 4x16:  lane<16 holds N=lane, VGPR0=K(k+0),VGPR1=K(k+1); lane>=16: K(k+2),K(k+3)
//   D 16x16: VGPR r: lane<16 -> (m0+r, n0+lane); lane>=16 -> (m0+8+r, n0+lane-16)
// ---------------------------------------------------------------------------
__global__ void gemm_wmma(const float* __restrict__ x, const float* __restrict__ mu,
                          float* __restrict__ out) {
    const int gwave = (int)((blockIdx.x * blockDim.x + threadIdx.x) >> 5);
    const int lane  = (int)(threadIdx.x & 31);
    // 512 tiles: 8 along M (batch), 64 along N (out features)
    const int tileN = gwave & 63;
    const int tileM = gwave >> 6;
    const int m0 = tileM * 16;
    const int n0 = tileN * 16;
    const int lm = lane & 15;   // row-in-half (A: M, B: N, D: N)
    const int kh = lane >> 4;   // half-wave selector -> k offset 2*kh

    const float* __restrict__ xrow = x  + (size_t)(m0 + lm) * NIN;  // A: fixed M, varying K
    const float* __restrict__ bcol = mu + (size_t)(n0 + lm);        // B: fixed N, row stride NOUT

    v8f c0 = {};
    v8f c1 = {};
    for (int k = 0; k < NIN; k += 8) {
        {
            int ka = k + 2 * kh;
            v2f a = *(const v2f*)(xrow + ka);
            v2f b;
            b.x = bcol[(size_t)(ka + 0) * NOUT];
            b.y = bcol[(size_t)(ka + 1) * NOUT];
            c0 = __builtin_amdgcn_wmma_f32_16x16x4_f32(false, a, false, b,
                                                       (short)0, c0, false, false);
        }
        {
            int ka = k + 4 + 2 * kh;
            v2f a = *(const v2f*)(xrow + ka);
            v2f b;
            b.x = bcol[(size_t)(ka + 0) * NOUT];
            b.y = bcol[(size_t)(ka + 1) * NOUT];
            c1 = __builtin_amdgcn_wmma_f32_16x16x4_f32(false, a, false, b,
                                                       (short)0, c1, false, false);
        }
    }
    v8f c = c0 + c1;

    const int mbase = m0 + (kh ? 8 : 0);
    float* __restrict__ o = out + (size_t)mbase * NOUT + n0 + lm;
#pragma unroll
    for (int r = 0; r < 8; ++r) {
        o[(size_t)r * NOUT] = c[r];
    }
}

// ---------------------------------------------------------------------------
// Kernel 3: the HBM-bound stream (sets the runtime: 512MB of eps @ 23.3TB/s).
//   out[b,o] += Sum_i (x[b,i]*eps[b,i,o]) * sigma[i,o]  +  sampled bias.
// Block = 128 threads: 16 o-quads (float4) x 8 batch rows. Grid 16x16 = 256 blocks.
// Grid is fixed at 1024 waves (one reduction per thread), so latency hiding
// must come from per-wave MLP: unroll 16 -> ~16 outstanding NT b128 eps loads
// per wave; 4 rotating accumulators keep the FMA chain off the critical path.
// eps is non-temporal (read-once, don't evict the 4MB sigma table from L2).
// ---------------------------------------------------------------------------
__global__ void fuse_kernel(const float* __restrict__ x,
                            const float* __restrict__ eps,
                            const float* __restrict__ ro,
                            const float* __restrict__ sigma_ws,
                            const float* __restrict__ mu_bias,
                            const float* __restrict__ ro_bias,
                            const float* __restrict__ eps_bias,
                            float* __restrict__ out,
                            int use_ws) {
    const int t  = (int)threadIdx.x;
    const int og = t & 15;          // which o-quad
    const int bl = t >> 4;          // 0..7 batch row within tile
    const int o0 = (int)blockIdx.x * 64 + og * 4;
    const int b  = (int)blockIdx.y * 8 + bl;

    const float* __restrict__ xr    = x   + (size_t)b * NIN;
    const float* __restrict__ ebase = eps + ((size_t)b * NIN) * NOUT + o0;

    v4f a0 = {0.f, 0.f, 0.f, 0.f};
    v4f a1 = {0.f, 0.f, 0.f, 0.f};
    v4f a2 = {0.f, 0.f, 0.f, 0.f};
    v4f a3 = {0.f, 0.f, 0.f, 0.f};

    if (use_ws) {
        const float* __restrict__ sbase = sigma_ws + o0;
        for (int i0 = 0; i0 < NIN; i0 += 16) {
            // Stage all 16 NT eps loads + 16 sigma loads, then FMA.
            v4f e[16], s[16];
            float xv[16];
#pragma unroll
            for (int u = 0; u < 16; ++u) {
                size_t row = (size_t)(i0 + u) * NOUT;
                e[u]  = __builtin_nontemporal_load((const v4f*)(ebase + row));
                s[u]  = *(const v4f*)(sbase + row);
                xv[u] = xr[i0 + u];
            }
#pragma unroll
            for (int u = 0; u < 16; u += 4) {
                a0 += (xv[u + 0] * e[u + 0]) * s[u + 0];
                a1 += (xv[u + 1] * e[u + 1]) * s[u + 1];
                a2 += (xv[u + 2] * e[u + 2]) * s[u + 2];
                a3 += (xv[u + 3] * e[u + 3]) * s[u + 3];
            }
        }
    } else {
        // Fallback: no workspace — compute softplus inline (trans-op heavy).
        const float* __restrict__ rbase = ro + o0;
#pragma unroll 4
        for (int i = 0; i < NIN; ++i) {
            float xv = xr[i];
            v4f e = __builtin_nontemporal_load((const v4f*)(ebase + (size_t)i * NOUT));
            v4f s = softplus4(*(const v4f*)(rbase + (size_t)i * NOUT));
            a0 += (xv * e) * s;
        }
    }

    v4f acc = (a0 + a1) + (a2 + a3);

    // bias = eps_bias * softplus(ro_bias) + mu_bias ; add GEMM partial from d_out
    v4f g  = *(const v4f*)(out      + (size_t)b * NOUT + o0);
    v4f sb = softplus4(*(const v4f*)(ro_bias + o0));
    v4f eb = *(const v4f*)(eps_bias + (size_t)b * NOUT + o0);
    v4f mb = *(const v4f*)(mu_bias  + o0);

    v4f res = g + acc + eb * sb + mb;
    *(v4f*)(out + (size_t)b * NOUT + o0) = res;
}

// ---------------------------------------------------------------------------
extern "C" void kernel_launch(void* const* d_in, const int* in_sizes, int n_in,
                              void* d_out, int out_size, void* d_ws, size_t ws_size,
                              hipStream_t stream) {
    const float* x        = (const float*)d_in[0];  // (128,1024)
    const float* mu       = (const float*)d_in[1];  // (1024,1024)
    const float* ro       = (const float*)d_in[2];  // (1024,1024)
    const float* mu_bias  = (const float*)d_in[3];  // (1,1024)
    const float* ro_bias  = (const float*)d_in[4];  // (1,1024)
    const float* eps      = (const float*)d_in[5];  // (128,1024,1024)
    const float* eps_bias = (const float*)d_in[6];  // (128,1024)
    float* out = (float*)d_out;                     // (128,1024)
    float* sigma = (float*)d_ws;

    const size_t sigma_bytes = (size_t)NIN * NOUT * sizeof(float);
    const int use_ws = (d_ws != nullptr && ws_size >= sigma_bytes) ? 1 : 0;

    if (use_ws) {
        // 1M elems / 4 per thread / 256 threads = 1024 blocks
        sigma_kernel<<<dim3((NIN * NOUT) / (4 * 256)), dim3(256), 0, stream>>>(ro, sigma);
    }

    // 512 tiles, 1 wave each, 8 waves (256 threads) per block -> 64 blocks
    gemm_wmma<<<dim3(64), dim3(256), 0, stream>>>(x, mu, out);

    // grid: 16 o-tiles (64 cols each) x 16 b-tiles (8 rows each); 128 threads
    fuse_kernel<<<dim3(NOUT / 64, NB / 8), dim3(128), 0, stream>>>(
        x, eps, ro, sigma, mu_bias, ro_bias, eps_bias, out, use_ws);
}